// IDWT_2D_10660108829257
// MI455X (gfx1250) — compile-verified
//
#include <hip/hip_runtime.h>

// IDWT 2D (Haar, 2x2) — pure streaming kernel for MI455X (gfx1250).
// Memory-bound: 268 MB total traffic @ 23.3 TB/s => ~11.5 us floor.
// Uses CDNA5 async global->LDS DMA (ASYNCcnt) for input staging and
// non-temporal stores (single-touch streams; don't pollute the 192 MB L2).

typedef float v2f __attribute__((ext_vector_type(2)));
typedef float v4f __attribute__((ext_vector_type(4)));
typedef int   b64v __attribute__((vector_size(2 * sizeof(int))));  // builtin's pointee type

#define AS1 __attribute__((address_space(1)))
#define AS3 __attribute__((address_space(3)))

#if defined(__has_builtin)
#  if __has_builtin(__builtin_amdgcn_global_load_async_to_lds_b64) && \
      __has_builtin(__builtin_amdgcn_s_wait_asynccnt)
#    define USE_ASYNC_LDS 1
#  else
#    define USE_ASYNC_LDS 0
#  endif
#else
#  define USE_ASYNC_LDS 0
#endif

__global__ __launch_bounds__(256) void idwt2d_haar_kernel(
    const float* __restrict__ x,   // (8, 4*64, 128, 128) f32
    float* __restrict__ y)         // (8, 64, 256, 256)   f32
{
    constexpr int W = 128;
    constexpr size_t inPlane  = (size_t)128 * 128;   // 16384
    constexpr size_t outPlane = inPlane * 4;          // 65536 (256*256)
    constexpr size_t kStride  = (size_t)64 * inPlane; // subband stride in channels

    const int t = threadIdx.x;
    const unsigned tid = blockIdx.x * 256u + (unsigned)t;

    // tid -> (n, c, i, jh): jh indexes an input column PAIR (2*jh, 2*jh+1)
    const int jh      = (int)(tid & 63u);
    const int i       = (int)((tid >> 6) & 127u);
    const unsigned nc = tid >> 13;             // n*64 + c, 0..511
    const int c       = (int)(nc & 63u);
    const int n       = (int)(nc >> 6);

    // input channel for subband k is n*256 + k*64 + c
    const size_t ibase = (((size_t)n * 256 + (size_t)c) * inPlane)
                       + (size_t)i * W + (size_t)jh * 2;

    v2f ll, lh, hl, hh;

#if USE_ASYNC_LDS
    // Stage the four subband b64 loads through LDS via the gfx1250 async DMA
    // path; each thread consumes only its own staged data, so a single
    // s_wait_asynccnt 0 is sufficient (no workgroup barrier needed).
    __shared__ v2f smem[4 * 256];
    __builtin_amdgcn_global_load_async_to_lds_b64(
        (AS1 b64v*)(x + ibase),               (AS3 b64v*)&smem[t],       0, 0);
    __builtin_amdgcn_global_load_async_to_lds_b64(
        (AS1 b64v*)(x + ibase + 1 * kStride), (AS3 b64v*)&smem[t + 256], 0, 0);
    __builtin_amdgcn_global_load_async_to_lds_b64(
        (AS1 b64v*)(x + ibase + 2 * kStride), (AS3 b64v*)&smem[t + 512], 0, 0);
    __builtin_amdgcn_global_load_async_to_lds_b64(
        (AS1 b64v*)(x + ibase + 3 * kStride), (AS3 b64v*)&smem[t + 768], 0, 0);
    __builtin_amdgcn_s_wait_asynccnt(0);
    ll = smem[t];
    lh = smem[t + 256];
    hl = smem[t + 512];
    hh = smem[t + 768];
#else
    ll = __builtin_nontemporal_load((const v2f*)(x + ibase));
    lh = __builtin_nontemporal_load((const v2f*)(x + ibase + 1 * kStride));
    hl = __builtin_nontemporal_load((const v2f*)(x + ibase + 2 * kStride));
    hh = __builtin_nontemporal_load((const v2f*)(x + ibase + 3 * kStride));
#endif

    // Haar butterfly:
    //   out[2i  ][2j] = LL+LH+HL+HH   out[2i  ][2j+1] = LL+LH-HL-HH
    //   out[2i+1][2j] = LL-LH+HL-HH   out[2i+1][2j+1] = LL-LH-HL+HH
    const v2f t0 = ll + lh;   // row a=0 partial
    const v2f t1 = ll - lh;   // row a=1 partial
    const v2f u0 = hl + hh;
    const v2f u1 = hl - hh;

    v4f r0, r1;
    r0.x = t0.x + u0.x;  r0.y = t0.x - u0.x;
    r0.z = t0.y + u0.y;  r0.w = t0.y - u0.y;
    r1.x = t1.x + u1.x;  r1.y = t1.x - u1.x;
    r1.z = t1.y + u1.y;  r1.w = t1.y - u1.y;

    // Output: y[n][c][2i + a][4*jh .. 4*jh+3] — lane-dense b128 NT stores.
    const size_t obase = (size_t)nc * outPlane + (size_t)(2 * i) * 256
                       + (size_t)jh * 4;
    __builtin_nontemporal_store(r0, (v4f*)(y + obase));
    __builtin_nontemporal_store(r1, (v4f*)(y + obase + 256));
}

extern "C" void kernel_launch(void* const* d_in, const int* in_sizes, int n_in,
                              void* d_out, int out_size, void* d_ws, size_t ws_size,
                              hipStream_t stream) {
    (void)in_sizes; (void)n_in; (void)out_size; (void)d_ws; (void)ws_size;
    const float* x = (const float*)d_in[0];
    // d_in[1] = filters: fixed Haar signs, folded into the kernel.
    float* y = (float*)d_out;

    constexpr unsigned total = 8u * 64u * 128u * 64u;  // 4,194,304 threads
    dim3 block(256);
    dim3 grid(total / 256u);                            // 16384 blocks
    idwt2d_haar_kernel<<<grid, block, 0, stream>>>(x, y);
}